// CustomPatchEmbedding_49263274885865
// MI455X (gfx1250) — compile-verified
//
#include <hip/hip_runtime.h>
#include <hip/hip_bf16.h>
#include <math.h>

typedef __attribute__((ext_vector_type(2))) float v2f;
typedef __attribute__((ext_vector_type(8))) float v8f;

// ---------------- workspace layout (ints) ----------------
// ws[0..3] : per-bucket record counts
// ws[4]    : S   (patches per row, from row 0)
// ws[5]    : total (per-row padded mask length)
// ws[6]    : M   (total record count)
// ws[7]    : spare
// ws[REC_OFF + 6*i .. +5] : record i = {row, slot, col, len, off, P}
// ws[IDX_OFF + k*MAXREC + j] : j-th record id of bucket k
#define MAXREC 65536
#define REC_OFF 8
#define IDX_OFF (REC_OFF + 6 * MAXREC)
#define BATCH 256
#define DMODEL 512

__global__ void init_kernel(int* ws) {
    if (threadIdx.x < 8) ws[threadIdx.x] = 0;
}

// One block per row. 256 threads.
__global__ void plan_kernel(const int* __restrict__ seg, int N, int* __restrict__ ws,
                            int P0, int P1, int P2, int P3) {
    const int b = blockIdx.x;
    const int* row = seg + (long)b * N;
    const int tid = threadIdx.x;

    __shared__ int sV;
    __shared__ int sCnt[256];
    __shared__ int sCols[2048];

    if (tid == 0) sV = N;
    __syncthreads();

    // first index where seg == -1 (valid length)
    int local = N;
    for (int i = tid; i < N; i += 256)
        if (row[i] == -1 && i < local) local = i;
    atomicMin(&sV, local);
    __syncthreads();
    const int V = sV;

    // chunked start detection + scan for ordered start list
    const int chunk = (N + 255) / 256;
    const int begin = tid * chunk;
    const int end   = min(begin + chunk, N);
    int cnt = 0;
    for (int i = begin; i < end; ++i)
        if (i < V && (i == 0 || row[i] != row[i - 1])) cnt++;
    sCnt[tid] = cnt;
    __syncthreads();
    // Hillis-Steele inclusive scan over 256 entries
    for (int ofs = 1; ofs < 256; ofs <<= 1) {
        int v = sCnt[tid];
        int add = (tid >= ofs) ? sCnt[tid - ofs] : 0;
        __syncthreads();
        sCnt[tid] = v + add;
        __syncthreads();
    }
    const int base = sCnt[tid] - cnt;   // exclusive prefix
    const int Sb = sCnt[255];           // segments in this row
    int j = base;
    for (int i = begin; i < end; ++i) {
        if (i < V && (i == 0 || row[i] != row[i - 1])) {
            if (j < 2048) sCols[j] = i;
            ++j;
        }
    }
    __syncthreads();

    if (tid == 0) {
        const int PL[4] = {P0, P1, P2, P3};
        if (b == 0) ws[4] = Sb;
        int off = 0;
        for (int s = 0; s < Sb && s < 2048; ++s) {
            const int c = sCols[s];
            const int L = ((s + 1 < Sb) ? sCols[s + 1] : V) - c;
            int best = 0, bd = abs(L - PL[0]);
            #pragma unroll
            for (int k = 1; k < 4; ++k) {
                int d = abs(L - PL[k]);
                if (d < bd) { bd = d; best = k; }
            }
            const int P = PL[best];
            int recId = atomicAdd(&ws[6], 1);
            if (recId < MAXREC) {
                int* rec = ws + REC_OFF + recId * 6;
                rec[0] = b; rec[1] = s; rec[2] = c; rec[3] = L; rec[4] = off; rec[5] = P;
                int li = atomicAdd(&ws[best], 1);
                if (li < MAXREC) ws[IDX_OFF + best * MAXREC + li] = recId;
            }
            off += P;
        }
        if (b == 0) ws[5] = off;
    }
}

// Persistent-wave grouped GEMM via V_WMMA_F32_16X16X4_F32.
// Each wave owns a fixed 16-column tile (B operand hoisted, loaded once), and
// strides over 16-patch tiles of its bucket. A-gathers are branchless
// (clamped unconditional loads + v_cndmask). Store epilogue is branchless:
// one ds_bpermute per output row (packed row|slot) and cos folded into sin.
// PC > 0: compile-time patch length (fully unrolled K loop). PC == 0: runtime p.
template <int PC>
__global__ void gemm_bucket_kernel(const float* __restrict__ x,
                                   const float* __restrict__ W,
                                   float* __restrict__ out,
                                   const int* __restrict__ ws,
                                   int N, int bucket, int p_rt) {
    const int P   = (PC > 0) ? PC : p_rt;
    const int NCH = (P + 3) >> 2;          // K chunks of 4
    const int lane = threadIdx.x & 31;
    const int hi   = lane >> 4;            // half-wave: K split
    const int wid  = blockIdx.x * (blockDim.x >> 5) + (threadIdx.x >> 5);
    const int nw   = gridDim.x * (blockDim.x >> 5);

    const int cntk = min(ws[bucket], MAXREC);
    if (cntk == 0) return;
    const int S   = ws[4];
    const int nPT = (cntk + 15) >> 4;

    const int ct = wid & 31;
    const int n  = ct * 16 + (lane & 15);  // output column for this lane
    const int ptStride = nw >> 5;

    // ---- hoisted B operand: W^T tile for this column, all K chunks ----
    v2f bfrag[(PC > 0) ? ((PC + 3) >> 2) : 6];
    #pragma unroll
    for (int c2 = 0; c2 < NCH; ++c2) {
        const int kb = c2 * 4 + (hi << 1);
        const float b0 = W[n * P + min(kb,     P - 1)];
        const float b1 = W[n * P + min(kb + 1, P - 1)];
        bfrag[c2].x = (kb     < P) ? b0 : 0.f;
        bfrag[c2].y = (kb + 1 < P) ? b1 : 0.f;
    }

    // positional embedding: pe(d, s) = sin(s*dt + phase), cos = sin(x + pi/2)
    const float dt    = __expf(-(float)(n & ~1) * (9.210340371976184f / (float)DMODEL));
    const float phase = (n & 1) ? 1.57079632679489662f : 0.0f;

    for (int pt = wid >> 5; pt < nPT; pt += ptStride) {
        const int m    = lane & 15;
        const int ridx = pt * 16 + m;
        int rr = 0, ss = 0, cc = 0, LL = 0;
        if (ridx < cntk) {
            const int recId = ws[IDX_OFF + bucket * MAXREC + ridx];
            const int* rec = ws + REC_OFF + recId * 6;
            rr = rec[0]; ss = rec[1]; cc = rec[2]; LL = rec[3];
        }
        const int pack    = (rr << 16) | (ss & 0xFFFF);   // row|slot for bpermute
        const int base    = rr * N + cc;                  // 0 for invalid records
        const int lim     = min(P, LL);                   // 0 for invalid -> all zeros
        const int clampHi = max(lim - 1, 0);

        v8f acc = {};
        #pragma unroll
        for (int c2 = 0; c2 < NCH; ++c2) {
            const int kb = c2 * 4 + (hi << 1);
            // branchless gather: clamped unconditional loads, zero via select
            float ax = x[base + min(kb,     clampHi)];
            float ay = x[base + min(kb + 1, clampHi)];
            v2f a;
            a.x = (kb     < lim) ? ax : 0.f;
            a.y = (kb + 1 < lim) ? ay : 0.f;
            acc = __builtin_amdgcn_wmma_f32_16x16x4_f32(
                    false, a, false, bfrag[c2], (short)0, acc, false, false);
        }

        // fused positional embedding + scatter store (C/D: vgpr g -> M = g + 8*hi)
        const int count = min(cntk - pt * 16, 16);
        #pragma unroll
        for (int g = 0; g < 8; ++g) {
            const int mm = g + (hi << 3);
            const int pk = __shfl(pack, mm, 32);          // one bpermute per row
            if (mm < count) {
                const int r_m = ((unsigned)pk) >> 16;
                const int s_m = pk & 0xFFFF;
                const float pe = __sinf((float)s_m * dt + phase);
                out[((long)(r_m * S + s_m)) * DMODEL + n] = acc[g] + pe;
            }
        }
    }
}

// One thread per record: write its P mask floats (1.0 = padding).
__global__ void mask_kernel(float* __restrict__ out, const int* __restrict__ ws) {
    const int M = min(ws[6], MAXREC);
    const int S = ws[4];
    const int total = ws[5];
    const long maskBase = (long)BATCH * S * DMODEL;
    for (int i = blockIdx.x * blockDim.x + threadIdx.x; i < M;
         i += gridDim.x * blockDim.x) {
        const int* rec = ws + REC_OFF + i * 6;
        const int r = rec[0], L = rec[3], off = rec[4], P = rec[5];
        float* dst = out + maskBase + (long)r * total + off;
        for (int t = 0; t < P; ++t) dst[t] = (t >= L) ? 1.0f : 0.0f;
    }
}

static void launch_bucket(const float* x, const float* W, float* out, const int* ws,
                          int N, int bucket, int P, hipStream_t stream) {
    dim3 g(256), b(256);
    switch (P) {
        case 5:  gemm_bucket_kernel<5 ><<<g, b, 0, stream>>>(x, W, out, ws, N, bucket, P); break;
        case 10: gemm_bucket_kernel<10><<<g, b, 0, stream>>>(x, W, out, ws, N, bucket, P); break;
        case 17: gemm_bucket_kernel<17><<<g, b, 0, stream>>>(x, W, out, ws, N, bucket, P); break;
        case 24: gemm_bucket_kernel<24><<<g, b, 0, stream>>>(x, W, out, ws, N, bucket, P); break;
        default: gemm_bucket_kernel<0 ><<<g, b, 0, stream>>>(x, W, out, ws, N, bucket, P); break;
    }
}

extern "C" void kernel_launch(void* const* d_in, const int* in_sizes, int n_in,
                              void* d_out, int out_size, void* d_ws, size_t ws_size,
                              hipStream_t stream) {
    const float* x   = (const float*)d_in[0];
    const int*   seg = (const int*)d_in[1];
    const float* Wk[4] = {(const float*)d_in[2], (const float*)d_in[3],
                          (const float*)d_in[4], (const float*)d_in[5]};
    float* out = (float*)d_out;
    int* ws = (int*)d_ws;

    const int N = in_sizes[0] / BATCH;   // 1376
    int P[4];
    for (int k = 0; k < 4; ++k) P[k] = in_sizes[2 + k] / DMODEL;  // 5,10,17,24

    init_kernel<<<1, 32, 0, stream>>>(ws);
    plan_kernel<<<BATCH, 256, 0, stream>>>(seg, N, ws, P[0], P[1], P[2], P[3]);
    for (int k = 0; k < 4; ++k)
        launch_bucket(x, Wk[k], out, ws, N, k, P[k], stream);
    mask_kernel<<<256, 256, 0, stream>>>(out, ws);
}